// LabelSmoothingCrossEntropyWithSuperKLDivLoss_35880156791216
// MI455X (gfx1250) — compile-verified
//
#include <hip/hip_runtime.h>
#include <math.h>

// ---------------------------------------------------------------------------
// LabelSmoothingCrossEntropy + Focal + SuperLoss (Lambert-W), forward only.
// Memory-bound: 412 MB single pass -> ~17.7 us floor @ 23.3 TB/s.
// CDNA5 path: async global->LDS pipeline (ASYNCcnt), depth 7, 8 LDS buffers.
// ---------------------------------------------------------------------------

#define TPB   256
#define NBUF  8            // LDS staging buffers (power of two)
#define DEPTH 7            // async tiles kept in flight

// ---- CDNA5 async global->LDS copy: 16 bytes per lane --------------------
__device__ __forceinline__ void async_copy16(unsigned lds_addr, unsigned goff,
                                             const float* sbase) {
  // GVS mode: mem_addr = SGPR64 + VGPR32 offset; vdst holds LDS byte address.
  asm volatile("global_load_async_to_lds_b128 %0, %1, %2"
               :
               : "v"(lds_addr), "v"(goff), "s"(sbase)
               : "memory");
}
#define WAIT_ASYNC(n) asm volatile("s_wait_asynccnt " #n ::: "memory")

// ---- online softmax accumulators ----------------------------------------
__device__ __forceinline__ void acc1(float v, float& m, float& s, float& t) {
  float nm = fmaxf(m, v);
  s = s * __expf(m - nm) + __expf(v - nm);
  t += v;
  m = nm;
}
__device__ __forceinline__ void acc4(float4 v, float& m, float& s, float& t) {
  float lm = fmaxf(fmaxf(v.x, v.y), fmaxf(v.z, v.w));
  float nm = fmaxf(m, lm);
  s = s * __expf(m - nm) + __expf(v.x - nm) + __expf(v.y - nm) +
      __expf(v.z - nm) + __expf(v.w - nm);
  t += v.x + v.y + v.z + v.w;
  m = nm;
}

__global__ __launch_bounds__(TPB) void row_loss_kernel(
    const float* __restrict__ x, const int* __restrict__ tgt,
    float* __restrict__ sl_out, float* __restrict__ rl_out, int C) {
  __shared__ float4 lbuf[NBUF][TPB];       // 32 KB staging ring
  __shared__ float redm[TPB / 32], reds[TPB / 32], redt[TPB / 32];

  const int row = blockIdx.x;
  const int tid = threadIdx.x;
  const float* p = x + (size_t)row * (size_t)C;

  float m = -INFINITY, s = 0.f, t = 0.f;

  // Peel leading scalars until 16B aligned (rows are only 4B aligned).
  const unsigned misb = (unsigned)((size_t)p & 15u);
  const int pre = (int)(((16u - misb) & 15u) >> 2);  // 0..3 floats
  if (tid < pre) acc1(p[tid], m, s, t);

  const float* pa = p + pre;               // 16B aligned
  const int nvec  = (C - pre) >> 2;        // float4 count
  const int rem   = (C - pre) & 3;         // trailing scalars
  const int ntile = nvec / TPB;            // full 256xfloat4 tiles
  const int tailv = nvec - ntile * TPB;

  // Partial tile: direct 128-bit loads.
  if (tid < tailv) {
    float4 v = ((const float4*)pa)[ntile * TPB + tid];
    acc4(v, m, s, t);
  }
  // Trailing scalars.
  if (tid < rem) acc1(pa[4 * nvec + tid], m, s, t);

  // Main stream: async global->LDS ring, DEPTH tiles in flight. Each lane
  // writes and reads only its own 16B slot, so s_wait_asynccnt alone orders
  // it (async loads complete in order within a wave) -- no barriers needed.
  const unsigned lbase = (unsigned)(size_t)&lbuf[0][tid];
  if (ntile > 0) {
    const int npro = (ntile < DEPTH) ? ntile : DEPTH;
    for (int i = 0; i < npro; ++i)
      async_copy16(lbase + (unsigned)((i & (NBUF - 1)) * (TPB * 16)),
                   (unsigned)(i * (TPB * 16) + tid * 16), pa);

    for (int t0 = 0; t0 < ntile; ++t0) {
      if (t0 + DEPTH < ntile) {
        const int ti = t0 + DEPTH;
        async_copy16(lbase + (unsigned)((ti & (NBUF - 1)) * (TPB * 16)),
                     (unsigned)(ti * (TPB * 16) + tid * 16), pa);
        WAIT_ASYNC(7);                     // tile t0 complete (in-order)
      } else {
        switch (ntile - 1 - t0) {          // drain: 6..0 outstanding
          case 6: WAIT_ASYNC(6); break;
          case 5: WAIT_ASYNC(5); break;
          case 4: WAIT_ASYNC(4); break;
          case 3: WAIT_ASYNC(3); break;
          case 2: WAIT_ASYNC(2); break;
          case 1: WAIT_ASYNC(1); break;
          default: WAIT_ASYNC(0); break;
        }
      }
      float4 v = lbuf[t0 & (NBUF - 1)][tid];
      acc4(v, m, s, t);
    }
  }

  // wave32 tree reduce of (max, scaled-sum, sum).
  for (int off = 16; off > 0; off >>= 1) {
    float om = __shfl_xor(m, off, 32);
    float os = __shfl_xor(s, off, 32);
    float ot = __shfl_xor(t, off, 32);
    float nm = fmaxf(m, om);
    s = s * __expf(m - nm) + os * __expf(om - nm);
    t += ot;
    m = nm;
  }
  const int wv = tid >> 5;
  if ((tid & 31) == 0) { redm[wv] = m; reds[wv] = s; redt[wv] = t; }
  __syncthreads();

  if (tid == 0) {
    float mm = redm[0], ss = reds[0], tsum = redt[0];
    for (int i = 1; i < TPB / 32; ++i) {
      float om = redm[i], os = reds[i], ot = redt[i];
      float nm = fmaxf(mm, om);
      ss = ss * __expf(mm - nm) + os * __expf(om - nm);
      tsum += ot;
      mm = nm;
    }
    const float lse = mm + logf(ss);
    const float rowloss = (float)C * lse - tsum;   // -sum(log_softmax)

    const int tg = tgt[row];
    const float xt = p[tg];
    const float ce = lse - xt;
    const float pt = expf(-ce);
    const float u = 1.f - pt;
    const float focal = 0.25f * u * u * ce;        // ALPHA=0.25, GAMMA=2

    const float tau = logf((float)C);
    const float beta = focal - tau;
    const float E = 2.718281828459045f;
    float y = 0.5f * fmaxf(-2.f / E, beta / 0.5f); // LAM=0.5
    y = fmaxf(y, -expf(-1.f) + 1e-7f);             // clamp to W0 domain
    float pq = sqrtf(2.f * (E * y + 1.f));
    float wbp = -1.f + pq - pq * pq * (1.f / 3.f);
    float w = (y < 0.f) ? wbp : log1pf(y);
#pragma unroll 1
    for (int it = 0; it < 20; ++it) {              // Halley, matches reference
      float ew = expf(w);
      float f = w * ew - y;
      float w1 = w + 1.f;
      w = w - f / (ew * w1 - (w + 2.f) * f / (2.f * w1));
    }
    const float sigma = expf(-w);                  // log(sigma) = -w
    const float sl = beta * sigma + 0.5f * w * w;  // LAM * log(sigma)^2

    sl_out[row] = sl;
    rl_out[row] = rowloss;
  }
}

__global__ __launch_bounds__(TPB) void finalize_kernel(
    const float* __restrict__ sl, const float* __restrict__ rl,
    float* __restrict__ out, int B, int C) {
  float a = 0.f, b = 0.f;
  for (int i = threadIdx.x; i < B; i += TPB) { a += sl[i]; b += rl[i]; }
  for (int off = 16; off > 0; off >>= 1) {
    a += __shfl_xor(a, off, 32);
    b += __shfl_xor(b, off, 32);
  }
  __shared__ float ra[TPB / 32], rb[TPB / 32];
  const int wv = threadIdx.x >> 5;
  if ((threadIdx.x & 31) == 0) { ra[wv] = a; rb[wv] = b; }
  __syncthreads();
  if (threadIdx.x == 0) {
    float sa = 0.f, sb = 0.f;
    for (int i = 0; i < TPB / 32; ++i) { sa += ra[i]; sb += rb[i]; }
    const float sl_mean = sa / (float)B;
    const float loss    = sb / (float)B;
    out[0] = loss * (0.01f / (float)C) + 0.99f * sl_mean;  // EPS=0.01
    out[1] = expf(sl_mean);
  }
}

extern "C" void kernel_launch(void* const* d_in, const int* in_sizes, int n_in,
                              void* d_out, int out_size, void* d_ws,
                              size_t ws_size, hipStream_t stream) {
  (void)n_in; (void)out_size; (void)ws_size;
  const float* x = (const float*)d_in[0];
  const int* tgt = (const int*)d_in[1];
  const int B = in_sizes[1];
  const int C = in_sizes[0] / B;

  float* ws = (float*)d_ws;
  float* sl = ws;        // [B] per-row SuperLoss
  float* rl = ws + B;    // [B] per-row -sum(log_softmax)

  row_loss_kernel<<<B, TPB, 0, stream>>>(x, tgt, sl, rl, C);
  finalize_kernel<<<1, TPB, 0, stream>>>(sl, rl, (float*)d_out, B, C);
}